// gat_conv_13606456394065
// MI455X (gfx1250) — compile-verified
//
#include <hip/hip_runtime.h>
#include <math.h>

// ---------------------------------------------------------------------------
// GATConv for MI455X (gfx1250, wave32).
//   N=50000 nodes, E=800000 edges, IN=256, H=4 heads, D=32 (H*D=128).
//   1) fh = feat @ W^T            -- fp32 WMMA 16x16x4 (full precision)
//   2) el/er + init emax/denom/rst(bias)
//   3) edge score + atomic float max  (segment max by dst)
//   4) exp + atomic add denom         (segment sum by dst)
//   5) wave-per-edge gather fh[src], scale, atomic scatter to rst[dst]
// ---------------------------------------------------------------------------

typedef __attribute__((ext_vector_type(2))) float v2f;
typedef __attribute__((ext_vector_type(8))) float v8f;

#define IN_F   256
#define HD     128   // H*D
#define NHEAD  4
#define LDA    260   // padded LDS row stride (floats): 4-float pad -> conflict-free

// ------------------------------- kernel 1 ----------------------------------
// One block = 16 nodes x 128 output cols. 8 waves, each owns a 16x16 tile.
__global__ __launch_bounds__(256)
void gat_proj_wmma(const float* __restrict__ feat,  // [N, 256]
                   const float* __restrict__ W,     // [128, 256]
                   float* __restrict__ fh,          // [N, 128]
                   int N) {
    __shared__ float sA[16 * LDA];                  // 16 rows x 256 (+pad)

    const int tid      = threadIdx.x;
    const int nodeBase = blockIdx.x * 16;

    // Cooperative load of the 16x256 feat tile (float4, fully coalesced).
    const float4* feat4 = (const float4*)feat;
    for (int i = tid; i < 16 * (IN_F / 4); i += 256) {
        int row = i >> 6;            // / 64
        int c4  = i & 63;            // % 64
        int gr  = nodeBase + row;
        if (gr >= N) gr = N - 1;     // clamp (N is a multiple of 16 anyway)
        float4 v = feat4[(size_t)gr * (IN_F / 4) + c4];
        *(float4*)&sA[row * LDA + c4 * 4] = v;
    }
    __syncthreads();

    const int lane    = tid & 31;
    const int wave    = tid >> 5;
    const int colBase = wave * 16;        // 8 waves cover all 128 cols
    const int mrow    = lane & 15;        // A: m index / B: n index
    const int khalf   = lane >> 4;        // 0 or 1 -> k offset 2*khalf

    // B[i][n] = W[n][i]  (fh = feat @ W^T): lane's column n = colBase+mrow,
    // so it streams along one row of W.
    const float* wrow = W + (size_t)(colBase + mrow) * IN_F;
    const float* arow = &sA[mrow * LDA];

    v8f acc = {};
    #pragma unroll 4
    for (int kk = 0; kk < IN_F; kk += 4) {
        const int k0 = kk + 2 * khalf;
        v2f a = *(const v2f*)&arow[k0];   // A 16x4 fp32 lane layout
        v2f b = *(const v2f*)&wrow[k0];   // B 4x16 fp32 lane layout
        acc = __builtin_amdgcn_wmma_f32_16x16x4_f32(
            /*neg_a=*/false, a, /*neg_b=*/false, b,
            /*c_mod=*/(short)0, acc, /*reuse_a=*/false, /*reuse_b=*/false);
    }

    // D layout: VGPR r, lane L -> M = r + 8*(L/16), N = L%16.
    float* out = fh + (size_t)nodeBase * HD + colBase + mrow;
    const int drow = 8 * khalf;
    if (nodeBase + 16 <= N) {
        // Full tile (all blocks when N % 16 == 0): unconditional store clause.
        #pragma unroll
        for (int r = 0; r < 8; ++r)
            out[(size_t)(drow + r) * HD] = acc[r];
    } else {
        #pragma unroll
        for (int r = 0; r < 8; ++r) {
            int m = drow + r;
            if (nodeBase + m < N) out[(size_t)m * HD] = acc[r];
        }
    }
}

// ------------------------------- kernel 2 ----------------------------------
// One wave per node: el/er dots, init emax/denom, seed rst with bias.
__global__ __launch_bounds__(256)
void gat_node_attn(const float* __restrict__ fh,
                   const float* __restrict__ attn_l,  // [4,32] flat
                   const float* __restrict__ attn_r,
                   const float* __restrict__ bias,    // [128]
                   float* __restrict__ el, float* __restrict__ er,
                   float* __restrict__ emax, float* __restrict__ denom,
                   float* __restrict__ rst, int N) {
    const int lane = threadIdx.x & 31;
    const int node = blockIdx.x * 8 + (threadIdx.x >> 5);
    if (node >= N) return;

    // lane covers cols c = 4*lane .. 4*lane+3  -> head h = lane>>3
    float4 v  = ((const float4*)(fh + (size_t)node * HD))[lane];
    float4 al = ((const float4*)attn_l)[lane];
    float4 ar = ((const float4*)attn_r)[lane];
    float pl = v.x * al.x + v.y * al.y + v.z * al.z + v.w * al.w;
    float pr = v.x * ar.x + v.y * ar.y + v.z * ar.z + v.w * ar.w;

    // Reduce within each group of 8 lanes (one head per group).
    #pragma unroll
    for (int off = 1; off < 8; off <<= 1) {
        pl += __shfl_xor(pl, off, 32);
        pr += __shfl_xor(pr, off, 32);
    }
    if ((lane & 7) == 0) {
        int h = lane >> 3;
        el[(size_t)node * NHEAD + h] = pl;
        er[(size_t)node * NHEAD + h] = pr;
    }
    if (lane < NHEAD) {
        emax [(size_t)node * NHEAD + lane] = -INFINITY;
        denom[(size_t)node * NHEAD + lane] = 0.0f;
    }
    // rst starts at bias (harness poisons d_out; re-seeded every call).
    ((float4*)(rst + (size_t)node * HD))[lane] = ((const float4*)bias)[lane];
}

// ---------------------------- float atomic max -----------------------------
__device__ __forceinline__ void atomicMaxF32(float* addr, float val) {
    // Valid because slots are initialized to -inf; order-preserving bit trick.
    if (val >= 0.0f) atomicMax((int*)addr, __float_as_int(val));
    else             atomicMin((unsigned int*)addr, __float_as_uint(val));
}

// ------------------------------- kernel 3 ----------------------------------
__global__ __launch_bounds__(256)
void gat_edge_score(const int* __restrict__ src, const int* __restrict__ dst,
                    const float* __restrict__ el, const float* __restrict__ er,
                    float* __restrict__ escore, float* __restrict__ emax, int E) {
    int e = blockIdx.x * blockDim.x + threadIdx.x;
    if (e >= E) return;
    int s = src[e], d = dst[e];
    float4 l4 = ((const float4*)el)[s];
    float4 r4 = ((const float4*)er)[d];
    float x[4] = {l4.x + r4.x, l4.y + r4.y, l4.z + r4.z, l4.w + r4.w};
    #pragma unroll
    for (int h = 0; h < NHEAD; ++h) {
        float v = x[h] > 0.0f ? x[h] : 0.2f * x[h];  // LeakyReLU(0.2)
        escore[(size_t)e * NHEAD + h] = v;
        atomicMaxF32(&emax[(size_t)d * NHEAD + h], v);
    }
}

// ------------------------------- kernel 4 ----------------------------------
__global__ __launch_bounds__(256)
void gat_edge_exp(const int* __restrict__ dst,
                  float* __restrict__ escore,       // in: e, out: exp(e-max)
                  const float* __restrict__ emax,
                  float* __restrict__ denom, int E) {
    int e = blockIdx.x * blockDim.x + threadIdx.x;
    if (e >= E) return;
    int d = dst[e];
    float4 m4 = ((const float4*)emax)[d];
    float mx[4] = {m4.x, m4.y, m4.z, m4.w};
    #pragma unroll
    for (int h = 0; h < NHEAD; ++h) {
        float t = __expf(escore[(size_t)e * NHEAD + h] - mx[h]);
        escore[(size_t)e * NHEAD + h] = t;
        atomicAdd(&denom[(size_t)d * NHEAD + h], t);
    }
}

// ------------------------------- kernel 5 ----------------------------------
// One wave per edge: 512B coalesced gather of fh[src], scaled atomic scatter.
__global__ __launch_bounds__(256)
void gat_edge_aggregate(const int* __restrict__ src, const int* __restrict__ dst,
                        const float* __restrict__ fh,
                        const float* __restrict__ escore,   // exp values
                        const float* __restrict__ denom,
                        float* __restrict__ rst, int E) {
    const int lane = threadIdx.x & 31;
    const int e    = blockIdx.x * 8 + (threadIdx.x >> 5);
    if (e >= E) return;
    const int s = src[e], d = dst[e];
    const int h = lane >> 3;                 // lane covers c = 4*lane..4*lane+3
    const float a = escore[(size_t)e * NHEAD + h] / denom[(size_t)d * NHEAD + h];
    float4 v = ((const float4*)(fh + (size_t)s * HD))[lane];
    float* o = rst + (size_t)d * HD + lane * 4;
    atomicAdd(o + 0, v.x * a);
    atomicAdd(o + 1, v.y * a);
    atomicAdd(o + 2, v.z * a);
    atomicAdd(o + 3, v.w * a);
}

// ---------------------------------------------------------------------------
extern "C" void kernel_launch(void* const* d_in, const int* in_sizes, int n_in,
                              void* d_out, int out_size, void* d_ws, size_t ws_size,
                              hipStream_t stream) {
    const float* feat   = (const float*)d_in[0];
    const int*   src    = (const int*)  d_in[1];
    const int*   dst    = (const int*)  d_in[2];
    const float* W      = (const float*)d_in[3];
    const float* attn_l = (const float*)d_in[4];
    const float* attn_r = (const float*)d_in[5];
    const float* bias   = (const float*)d_in[6];
    float*       out    = (float*)d_out;

    const int N = in_sizes[0] / IN_F;   // 50000
    const int E = in_sizes[1];          // 800000

    // Workspace carve-up (floats).
    float* ws     = (float*)d_ws;
    float* fh     = ws;                          // N*128
    float* el     = fh    + (size_t)N * HD;      // N*4
    float* er     = el    + (size_t)N * NHEAD;   // N*4
    float* emax   = er    + (size_t)N * NHEAD;   // N*4
    float* denom  = emax  + (size_t)N * NHEAD;   // N*4
    float* escore = denom + (size_t)N * NHEAD;   // E*4

    // 1) projection GEMM (WMMA fp32)
    gat_proj_wmma<<<dim3((N + 15) / 16), dim3(256), 0, stream>>>(feat, W, fh, N);
    // 2) per-node logits + inits
    gat_node_attn<<<dim3((N + 7) / 8), dim3(256), 0, stream>>>(
        fh, attn_l, attn_r, bias, el, er, emax, denom, out, N);
    // 3) edge scores + segment max
    gat_edge_score<<<dim3((E + 255) / 256), dim3(256), 0, stream>>>(
        src, dst, el, er, escore, emax, E);
    // 4) exp + segment sum
    gat_edge_exp<<<dim3((E + 255) / 256), dim3(256), 0, stream>>>(
        dst, escore, emax, denom, E);
    // 5) weighted aggregation
    gat_edge_aggregate<<<dim3((E + 7) / 8), dim3(256), 0, stream>>>(
        src, dst, fh, escore, denom, out, E);
}